// WarpDTI_64622077935985
// MI455X (gfx1250) — compile-verified
//
#include <hip/hip_runtime.h>
#include <math.h>

#define NB 2
#define NH 128
#define NW 128
#define ND 128
#define HWD (NH * NW * ND)

typedef int v4i_t __attribute__((ext_vector_type(4)));
typedef __attribute__((address_space(1))) v4i_t g_v4i;   // global
typedef __attribute__((address_space(3))) v4i_t l_v4i;   // LDS

__device__ __forceinline__ void async_copy16(const float* g, float* l) {
#if __has_builtin(__builtin_amdgcn_global_load_async_to_lds_b128)
    // CDNA5 async global->LDS DMA (per-lane, tracked by ASYNCcnt)
    __builtin_amdgcn_global_load_async_to_lds_b128((g_v4i*)g, (l_v4i*)l, 0, 0);
#else
    *(float4*)l = *(const float4*)g;  // synchronous fallback
#endif
}

__device__ __forceinline__ void wait_async0() {
#if __has_builtin(__builtin_amdgcn_s_wait_asynccnt)
    __builtin_amdgcn_s_wait_asynccnt(0);
#else
    asm volatile("s_wait_asynccnt 0" ::: "memory");
#endif
}

__global__ __launch_bounds__(128) void warp_dti_kernel(
    const float* __restrict__ dti,   // (B, 6, H, W, D)
    const float* __restrict__ ddf,   // (B, 3, H, W, D)
    float* __restrict__ out)         // (B, 6, H, W, D)
{
    // 15 z-lines: [comp*5 + which][z], which: 0=(x,y) 1=(x-1,y) 2=(x+1,y) 3=(x,y-1) 4=(x,y+1)
    __shared__ float sh[15][ND];

    const int z = threadIdx.x;
    int lin = blockIdx.x;
    const int y = lin % NW; lin /= NW;
    const int x = lin % NH; lin /= NH;
    const int b = lin;

    const int xm = max(x - 1, 0), xp = min(x + 1, NH - 1);
    const int ym = max(y - 1, 0), yp = min(y + 1, NW - 1);

    // ---- Stage ddf lines into LDS via async tensor-path copies -------------
    // 15 lines * 32 chunks of 16B = 480 chunks, 128 threads
    for (int c = threadIdx.x; c < 15 * (ND / 4); c += blockDim.x) {
        const int l = c >> 5;             // line index (32 chunks per line)
        const int e = (c & 31) << 2;      // element offset inside line
        const int comp  = l / 5;
        const int which = l % 5;
        const int lx = (which == 1) ? xm : (which == 2) ? xp : x;
        const int ly = (which == 3) ? ym : (which == 4) ? yp : y;
        const float* g = ddf + (((size_t)(b * 3 + comp) * NH + lx) * NW + ly) * ND + e;
        async_copy16(g, &sh[l][e]);
    }
    wait_async0();
    __syncthreads();

    // ---- Jacobian J = I + grad(ddf), one-sided at edges --------------------
    const int zm = max(z - 1, 0), zp = min(z + 1, ND - 1);
    const float sxw = 1.0f / (float)(xp - xm);   // 1 at edges, 0.5 interior
    const float syw = 1.0f / (float)(yp - ym);
    const float szw = 1.0f / (float)(zp - zm);

    float J[3][3];
    float u[3];
#pragma unroll
    for (int cmp = 0; cmp < 3; ++cmp) {
        const float* L = sh[cmp * 5 + 0];
        u[cmp]    = L[z];
        J[cmp][0] = (sh[cmp * 5 + 2][z] - sh[cmp * 5 + 1][z]) * sxw;
        J[cmp][1] = (sh[cmp * 5 + 4][z] - sh[cmp * 5 + 3][z]) * syw;
        J[cmp][2] = (L[zp] - L[zm]) * szw;
    }
    J[0][0] += 1.0f; J[1][1] += 1.0f; J[2][2] += 1.0f;

    // ---- Trilinear warp of the 6 DTI channels (border clamp) ---------------
    float px = fminf(fmaxf(u[0] + (float)x, 0.0f), (float)(NH - 1));
    float py = fminf(fmaxf(u[1] + (float)y, 0.0f), (float)(NW - 1));
    float pz = fminf(fmaxf(u[2] + (float)z, 0.0f), (float)(ND - 1));
    float fx0 = floorf(px), fy0 = floorf(py), fz0 = floorf(pz);
    float fx = px - fx0, fy = py - fy0, fz = pz - fz0;
    int x0i = (int)fx0, y0i = (int)fy0, z0i = (int)fz0;
    int x1i = min(x0i + 1, NH - 1);
    int y1i = min(y0i + 1, NW - 1);
    int z1i = min(z0i + 1, ND - 1);

    const int i000 = (x0i * NW + y0i) * ND + z0i;
    const int i001 = (x0i * NW + y0i) * ND + z1i;
    const int i010 = (x0i * NW + y1i) * ND + z0i;
    const int i011 = (x0i * NW + y1i) * ND + z1i;
    const int i100 = (x1i * NW + y0i) * ND + z0i;
    const int i101 = (x1i * NW + y0i) * ND + z1i;
    const int i110 = (x1i * NW + y1i) * ND + z0i;
    const int i111 = (x1i * NW + y1i) * ND + z1i;

    const float gx = 1.0f - fx, gy = 1.0f - fy, gz = 1.0f - fz;
    const float w000 = gx * gy * gz, w001 = gx * gy * fz;
    const float w010 = gx * fy * gz, w011 = gx * fy * fz;
    const float w100 = fx * gy * gz, w101 = fx * gy * fz;
    const float w110 = fx * fy * gz, w111 = fx * fy * fz;

    const float* img = dti + (size_t)b * 6 * HWD;
    float Dch[6];
#pragma unroll
    for (int ch = 0; ch < 6; ++ch) {
        const float* p = img + (size_t)ch * HWD;
        Dch[ch] = w000 * p[i000] + w001 * p[i001]
                + w010 * p[i010] + w011 * p[i011]
                + w100 * p[i100] + w101 * p[i101]
                + w110 * p[i110] + w111 * p[i111];
    }

    // ---- Orthogonal polar factor R = U Vh = J (J^T J)^{-1/2} ---------------
    // Determinant-scaled Newton: X <- 0.5 * (z*X + (1/z)*X^{-T}), z=|det|^(-1/3)
    float X[3][3];
#pragma unroll
    for (int i = 0; i < 3; ++i)
#pragma unroll
        for (int j = 0; j < 3; ++j) X[i][j] = J[i][j];

#pragma unroll
    for (int it = 0; it < 6; ++it) {
        float C[3][3];   // cofactor matrix: X^{-T} = C / det
        C[0][0] = X[1][1] * X[2][2] - X[1][2] * X[2][1];
        C[0][1] = X[1][2] * X[2][0] - X[1][0] * X[2][2];
        C[0][2] = X[1][0] * X[2][1] - X[1][1] * X[2][0];
        C[1][0] = X[0][2] * X[2][1] - X[0][1] * X[2][2];
        C[1][1] = X[0][0] * X[2][2] - X[0][2] * X[2][0];
        C[1][2] = X[0][1] * X[2][0] - X[0][0] * X[2][1];
        C[2][0] = X[0][1] * X[1][2] - X[0][2] * X[1][1];
        C[2][1] = X[0][2] * X[1][0] - X[0][0] * X[1][2];
        C[2][2] = X[0][0] * X[1][1] - X[0][1] * X[1][0];
        float det = X[0][0] * C[0][0] + X[0][1] * C[0][1] + X[0][2] * C[0][2];
        float ad  = fmaxf(fabsf(det), 1e-30f);
        det = (det < 0.0f) ? -ad : ad;
        const float zs = exp2f(-(1.0f / 3.0f) * log2f(ad));  // |det|^(-1/3)
        const float a  = 0.5f * zs;
        const float t  = 0.5f / (zs * det);
#pragma unroll
        for (int i = 0; i < 3; ++i)
#pragma unroll
            for (int j = 0; j < 3; ++j)
                X[i][j] = a * X[i][j] + t * C[i][j];
    }

    // ---- O = R^T * D * R ----------------------------------------------------
    const float Dm[3][3] = {{Dch[0], Dch[1], Dch[3]},
                            {Dch[1], Dch[2], Dch[4]},
                            {Dch[3], Dch[4], Dch[5]}};
    float M[3][3];  // M = Dm * R
#pragma unroll
    for (int i = 0; i < 3; ++i)
#pragma unroll
        for (int j = 0; j < 3; ++j)
            M[i][j] = Dm[i][0] * X[0][j] + Dm[i][1] * X[1][j] + Dm[i][2] * X[2][j];
    float O[3][3];  // O = R^T * M
#pragma unroll
    for (int i = 0; i < 3; ++i)
#pragma unroll
        for (int j = 0; j < 3; ++j)
            O[i][j] = X[0][i] * M[0][j] + X[1][i] * M[1][j] + X[2][i] * M[2][j];

    // ---- lower-triangle channels: (0,0)(1,0)(1,1)(2,0)(2,1)(2,2) -----------
    const size_t obase = (size_t)b * 6 * HWD + ((size_t)x * NW + y) * ND + z;
    out[obase + 0 * (size_t)HWD] = O[0][0];
    out[obase + 1 * (size_t)HWD] = O[1][0];
    out[obase + 2 * (size_t)HWD] = O[1][1];
    out[obase + 3 * (size_t)HWD] = O[2][0];
    out[obase + 4 * (size_t)HWD] = O[2][1];
    out[obase + 5 * (size_t)HWD] = O[2][2];
}

extern "C" void kernel_launch(void* const* d_in, const int* in_sizes, int n_in,
                              void* d_out, int out_size, void* d_ws, size_t ws_size,
                              hipStream_t stream) {
    (void)in_sizes; (void)n_in; (void)out_size; (void)d_ws; (void)ws_size;
    const float* dti = (const float*)d_in[0];
    const float* ddf = (const float*)d_in[1];
    float* out = (float*)d_out;
    warp_dti_kernel<<<dim3(NB * NH * NW), dim3(128), 0, stream>>>(dti, ddf, out);
}